// ContrastiveLoss_29944511987848
// MI455X (gfx1250) — compile-verified
//
#include <hip/hip_runtime.h>
#include <hip/hip_bf16.h>
#include <math.h>
#include <stdint.h>

typedef __attribute__((ext_vector_type(2))) float v2f;
typedef __attribute__((ext_vector_type(8))) float v8f;

#define N_BANK 444
#define TT 64
#define DD 512
#define BB 222
#define K_TOT 54
#define EPSF 1e-8f

// ---------------------------------------------------------------------------
// Kernel 1: per-row squared norms: norm2[n*64+t] = sum_d emb[n][t][d]^2
// One wave32 per (n,t) row; coalesced strided loads + shuffle reduce.
// ---------------------------------------------------------------------------
__global__ void norms_kernel(const float* __restrict__ emb, float* __restrict__ norm2) {
    int gw   = (blockIdx.x * blockDim.x + threadIdx.x) >> 5;   // global wave id = row id
    int lane = threadIdx.x & 31;
    if (gw >= N_BANK * TT) return;
    const float* p = emb + (size_t)gw * DD;
    float s = 0.f;
    for (int d = lane; d < DD; d += 32) { float v = p[d]; s = fmaf(v, v, s); }
#pragma unroll
    for (int off = 16; off > 0; off >>= 1) s += __shfl_xor(s, off, 32);
    if (lane == 0) norm2[gw] = s;
}

// ---------------------------------------------------------------------------
// Kernel 2: zero the pos/neg accumulators (they are contiguous in workspace)
// ---------------------------------------------------------------------------
__global__ void zero_kernel(float* __restrict__ p, int n) {
    int i = blockIdx.x * blockDim.x + threadIdx.x;
    if (i < n) p[i] = 0.f;
}

// ---------------------------------------------------------------------------
// Kernel 3: main WMMA kernel. One block per (b,k). Computes
//   maxnorm[b,k,t] = max(exp(-min_s d2[t,s]), EPS)
// and atomically accumulates into possum/negsum[b*64+t].
// Double-buffered LDS fed by CDNA5 async global->LDS loads (ASYNCcnt).
// ---------------------------------------------------------------------------
__launch_bounds__(256)
__global__ void qc_kernel(const float* __restrict__ emb,
                          const int*   __restrict__ posself,
                          const int*   __restrict__ poscross,
                          const int*   __restrict__ neg,
                          const int*   __restrict__ order,
                          const float* __restrict__ norm2,
                          float*       __restrict__ possum,
                          float*       __restrict__ negsum) {
    __shared__ float    qs[2][64][68];   // padded stride: conflict-free fragment reads
    __shared__ float    cs[2][64][68];
    __shared__ float    q2s[64];
    __shared__ float    c2s[64];
    __shared__ unsigned redU[64];        // running min of d2 per t (uint bits; d2 >= 0)

    const int bk = blockIdx.x;
    const int b  = bk / K_TOT;
    const int k  = bk - b * K_TOT;

    // combined = [pos_idx(12) | neg(30) | order[pos_idx](12)]
    int idx;
    if (k < 12) {
        idx = (k < 6) ? posself[b * 6 + k] : poscross[b * 6 + (k - 6)];
    } else if (k < 42) {
        idx = neg[b * 30 + (k - 12)];
    } else {
        int kk2 = k - 42;
        int p   = (kk2 < 6) ? posself[b * 6 + kk2] : poscross[b * 6 + (kk2 - 6)];
        idx = order[p];
    }

    const int tid = threadIdx.x;
    if (tid < 64) {
        q2s[tid]  = norm2[b   * TT + tid];
        c2s[tid]  = norm2[idx * TT + tid];
        redU[tid] = 0x7F800000u;   // +inf bit pattern
    }

    const float* qg = emb + (size_t)b   * TT * DD;
    const float* cg = emb + (size_t)idx * TT * DD;

    const int lane  = tid & 31;
    const int w     = tid >> 5;                 // wave id 0..7
    const int mA    = lane & 15;                // A/B fragment row/col index
    const int halfk = (lane >> 4) << 1;         // K sub-offset per half-wave
    // wave w owns output tiles w and w+8 of the 4x4 tiling of the 64x64 result.
    // (w+8)&3 == w&3, so both tiles share the same B (column) fragments.
    const int ti0 = ((w)     >> 2) * 16, tj0 = ((w)     & 3) * 16;
    const int ti1 = ((w + 8) >> 2) * 16, tj1 = ((w + 8) & 3) * 16;

    // Per-thread staging slots: 4 x float4 per array per chunk.
    int rowS[4], colS[4];
#pragma unroll
    for (int i = 0; i < 4; ++i) {
        int lin = tid + 256 * i;
        rowS[i] = lin >> 4;
        colS[i] = (lin & 15) << 2;
    }

    // Issue async global->LDS loads for chunk dc into buffer buf.
    auto stage = [&](int dc, int buf) {
#pragma unroll
        for (int i = 0; i < 4; ++i) {
            unsigned qoff = (unsigned)(uintptr_t)&qs[buf][rowS[i]][colS[i]];
            unsigned coff = (unsigned)(uintptr_t)&cs[buf][rowS[i]][colS[i]];
            uint64_t qa = (uint64_t)(qg + (size_t)rowS[i] * DD + dc * 64 + colS[i]);
            uint64_t ca = (uint64_t)(cg + (size_t)rowS[i] * DD + dc * 64 + colS[i]);
            asm volatile("global_load_async_to_lds_b128 %0, %1, off"
                         :: "v"(qoff), "v"(qa) : "memory");
            asm volatile("global_load_async_to_lds_b128 %0, %1, off"
                         :: "v"(coff), "v"(ca) : "memory");
        }
    };

    v8f acc0 = {0.f,0.f,0.f,0.f,0.f,0.f,0.f,0.f};
    v8f acc1 = {0.f,0.f,0.f,0.f,0.f,0.f,0.f,0.f};

    stage(0, 0);   // prologue: chunk 0 -> buffer 0

    for (int dc = 0; dc < 8; ++dc) {
        const int cur = dc & 1;
        // My async stores for buf[cur] done, then block-wide visibility.
        asm volatile("s_wait_asynccnt 0x0" ::: "memory");
        __syncthreads();
        // Prefetch next chunk into the other buffer (safe: everyone finished
        // reading it in iteration dc-1 before reaching the barrier above).
        if (dc + 1 < 8) stage(dc + 1, cur ^ 1);

        // 16 WMMA k-steps of 4 over this 64-wide chunk, two tiles per wave.
#pragma unroll
        for (int kk = 0; kk < 16; ++kk) {
            const int ko = (kk << 2) + halfk;
            v2f a0 = *(const v2f*)&qs[cur][ti0 + mA][ko];
            v2f b0 = *(const v2f*)&cs[cur][tj0 + mA][ko];
            acc0 = __builtin_amdgcn_wmma_f32_16x16x4_f32(false, a0, false, b0,
                                                         (short)0, acc0, false, false);
            v2f a1 = *(const v2f*)&qs[cur][ti1 + mA][ko];
            v2f b1 = *(const v2f*)&cs[cur][tj1 + mA][ko];
            acc1 = __builtin_amdgcn_wmma_f32_16x16x4_f32(false, a1, false, b1,
                                                         (short)0, acc1, false, false);
        }
    }

    // Epilogue: d2 = q2[t] + c2[s] - 2*qc, clamp >= 0, min over s.
    // C/D layout: VGPR r, lanes 0-15 -> M=r, lanes 16-31 -> M=8+r; N = lane%16.
    {
        const int n  = lane & 15;
        const int mh = (lane >> 4) << 3;

        float c2v = c2s[tj0 + n];
#pragma unroll
        for (int r = 0; r < 8; ++r) {
            float d2 = fmaxf(q2s[ti0 + mh + r] + c2v - 2.0f * acc0[r], 0.0f);
#pragma unroll
            for (int off = 8; off > 0; off >>= 1) d2 = fminf(d2, __shfl_xor(d2, off, 32));
            if (n == 0) atomicMin(&redU[ti0 + mh + r], __float_as_uint(d2));
        }
        c2v = c2s[tj1 + n];
#pragma unroll
        for (int r = 0; r < 8; ++r) {
            float d2 = fmaxf(q2s[ti1 + mh + r] + c2v - 2.0f * acc1[r], 0.0f);
#pragma unroll
            for (int off = 8; off > 0; off >>= 1) d2 = fminf(d2, __shfl_xor(d2, off, 32));
            if (n == 0) atomicMin(&redU[ti1 + mh + r], __float_as_uint(d2));
        }
    }
    __syncthreads();

    if (tid < 64) {
        float dmin = __uint_as_float(redU[tid]);
        float val  = fmaxf(expf(-dmin), EPSF);
        float* dst = (k < 12) ? possum : negsum;
        atomicAdd(&dst[b * TT + tid], val);
    }
}

// ---------------------------------------------------------------------------
// Kernel 4: final scalar loss reduction
// ---------------------------------------------------------------------------
__global__ void finalize_kernel(const float* __restrict__ possum,
                                const float* __restrict__ negsum,
                                float* __restrict__ out) {
    __shared__ float red[256];
    int tid = threadIdx.x;
    float s = 0.f;
    for (int i = tid; i < BB * TT; i += 256) {
        float p = possum[i], ng = negsum[i];
        s += -logf(p / (p + ng + EPSF));
    }
    red[tid] = s;
    __syncthreads();
    for (int st = 128; st > 0; st >>= 1) {
        if (tid < st) red[tid] += red[tid + st];
        __syncthreads();
    }
    if (tid == 0) out[0] = 500.0f * red[0] / (float)BB;
}

// ---------------------------------------------------------------------------
extern "C" void kernel_launch(void* const* d_in, const int* in_sizes, int n_in,
                              void* d_out, int out_size, void* d_ws, size_t ws_size,
                              hipStream_t stream) {
    const float* emb      = (const float*)d_in[0];
    const int*   posself  = (const int*)d_in[1];
    const int*   poscross = (const int*)d_in[2];
    const int*   neg      = (const int*)d_in[3];
    const int*   order    = (const int*)d_in[4];
    float*       out      = (float*)d_out;

    // Workspace layout (floats): norm2[444*64] | possum[222*64] | negsum[222*64]
    float* norm2  = (float*)d_ws;
    float* possum = norm2 + N_BANK * TT;
    float* negsum = possum + BB * TT;

    // 1) per-row norms: 444*64 rows, one wave each, 8 waves per block
    {
        int rows = N_BANK * TT;                 // 28416
        int blocks = (rows + 7) / 8;            // 3552
        norms_kernel<<<blocks, 256, 0, stream>>>(emb, norm2);
    }
    // 2) zero pos/neg accumulators (contiguous)
    {
        int n = 2 * BB * TT;                    // 28416
        zero_kernel<<<(n + 255) / 256, 256, 0, stream>>>(possum, n);
    }
    // 3) main WMMA kernel: one block per (b,k)
    qc_kernel<<<BB * K_TOT, 256, 0, stream>>>(emb, posself, poscross, neg, order,
                                              norm2, possum, negsum);
    // 4) final loss
    finalize_kernel<<<1, 256, 0, stream>>>(possum, negsum, out);
}